// SelfAttentionLayer_V2_73065983640153
// MI455X (gfx1250) — compile-verified
//
#include <hip/hip_runtime.h>
#include <hip/hip_bf16.h>

// Problem constants (match reference)
#define B_SZ   512
#define M_SZ   4096
#define N_SZ   64
#define K_SZ   64
#define H_SZ   8
#define DK_SZ  32
#define DV_SZ  4
#define KER_SZ 3
#define DOUT_PER 64

typedef __attribute__((ext_vector_type(2))) float v2f;
typedef __attribute__((ext_vector_type(8))) float v8f;

// D = A(16x4, fp32) * B(4x16, fp32) + C(16x16, fp32)
__device__ __forceinline__ v8f wmma_k4(v2f a, v2f b, v8f c) {
  // 8 args: (neg_a, A, neg_b, B, c_mod, C, reuse_a, reuse_b)
  return __builtin_amdgcn_wmma_f32_16x16x4_f32(
      false, a, false, b, (short)0, c, false, false);
}

// One 16x16 output tile, K accumulated in steps of 4.
// A layout: lane<16 -> row m=lane, K pair {k0,k0+1}; lane>=16 -> same rows, K pair {k0+2,k0+3}
// B layout mirrors A with columns on lanes.
template <int KTOT, typename FA, typename FB>
__device__ __forceinline__ v8f tile_gemm(int lane, FA loadA, FB loadB, v8f acc) {
  const int mn   = lane & 15;
  const int koff = (lane >> 4) << 1;
#pragma unroll
  for (int k0 = 0; k0 < KTOT; k0 += 4) {
    const int k = k0 + koff;
    v2f a, b;
    a.x = loadA(mn, k);
    a.y = loadA(mn, k + 1);
    b.x = loadB(k, mn);
    b.y = loadB(k + 1, mn);
    acc = wmma_k4(a, b, acc);
  }
  return acc;
}

// C/D layout: VGPR r, lanes 0-15 -> (m=r, n=lane); lanes 16-31 -> (m=r+8, n=lane-16)
template <typename FS>
__device__ __forceinline__ void tile_store(int lane, v8f acc, FS st) {
  const int n   = lane & 15;
  const int mhi = (lane >> 4) << 3;
#pragma unroll
  for (int r = 0; r < 8; ++r) st(mhi + r, n, acc[r]);
}

// LDS layout (floats), padded row pitches to dodge bank conflicts on A-operand reads
#define OFF_FEAT 0                    // 4096
#define OFF_QIN  4096                 // 64*65 = 4160
#define OFF_KF   (4096 + 4160)        // 64*65 = 4160  (kf -> v -> O, reused)
#define OFF_QH   (OFF_KF + 4160)      // 64*33 = 2112
#define OFF_KEY  (OFF_QH + 2112)      // 32*64 = 2048
#define OFF_SBUF (OFF_KEY + 2048)     // 64*65 = 4160
#define OFF_RED  (OFF_SBUF + 4160)    // 512
#define SMEM_FLOATS (OFF_RED + 512)   // 21248 floats = 84992 bytes

__global__ __launch_bounds__(256) void fused_attn_kernel(
    const float* __restrict__ features,   // (B, M)
    const int*   __restrict__ sidx,       // (H, M)
    const float* __restrict__ gamma,      // (M)
    const float* __restrict__ beta,       // (M)
    const float* __restrict__ Wq,         // (H*DK, K)
    const float* __restrict__ Wk,         // (H, DK, K)
    const float* __restrict__ Wv1,        // (H, DV, KER)
    const float* __restrict__ Wv2,        // (H, DV)
    const float* __restrict__ Wfc,        // (DOUT_PER, H*K)
    float* __restrict__ out)              // [B*M out] ++ [B*H*N*N attn]
{
  extern __shared__ float smem[];
  float* feat = smem + OFF_FEAT;
  float* qin  = smem + OFF_QIN;
  float* kf   = smem + OFF_KF;
  float* qh   = smem + OFF_QH;
  float* keyb = smem + OFF_KEY;
  float* sbuf = smem + OFF_SBUF;
  float* red  = smem + OFF_RED;

  const int t    = threadIdx.x;
  const int lane = t & 31;
  const int wv   = t >> 5;          // wave id 0..7
  const int b    = blockIdx.x;
  const float* fr = features + (size_t)b * M_SZ;

  // ---- Phase 0: stage features row, LayerNorm into qin (pitch 65) ----
#pragma unroll
  for (int i = 0; i < 16; ++i) feat[t + i * 256] = fr[t + i * 256];
  __syncthreads();

  float s1 = 0.f, s2 = 0.f;
#pragma unroll
  for (int i = 0; i < 16; ++i) { float x = feat[t + i * 256]; s1 += x; s2 += x * x; }
  red[t] = s1; red[256 + t] = s2;
  __syncthreads();
  for (int s = 128; s > 0; s >>= 1) {
    if (t < s) { red[t] += red[t + s]; red[256 + t] += red[256 + t + s]; }
    __syncthreads();
  }
  const float mu   = red[0] * (1.0f / M_SZ);
  const float var  = red[256] * (1.0f / M_SZ) - mu * mu;
  const float rstd = rsqrtf(var + 1e-5f);
#pragma unroll
  for (int i = 0; i < 16; ++i) {
    int j = t + i * 256;
    qin[(j >> 6) * 65 + (j & 63)] = (feat[j] - mu) * rstd * gamma[j] + beta[j];
  }

  // persistent FC accumulators: wave wv owns output tiles (2*wv) and (2*wv+1)
  v8f facc0 = {};
  v8f facc1 = {};

  for (int h = 0; h < H_SZ; ++h) {
    __syncthreads();  // also protects kf reuse vs previous head's FC reads

    // ---- gather kf_h[n][k] = feat[idx[h][n*64+k]]  (pitch 65) ----
    const int* idxh = sidx + h * M_SZ;
#pragma unroll
    for (int i = 0; i < 16; ++i) {
      int j = t + i * 256;
      kf[(j >> 6) * 65 + (j & 63)] = feat[idxh[j]];
    }
    __syncthreads();

    // ---- q_h = qin @ Wq[h*32 : h*32+32]^T  (64x32), 8 tiles, 1/wave ----
    {
      const int mb = (wv >> 1) << 4;      // n-tile
      const int eb = (wv & 1) << 4;       // dk-tile
      const float* WqH = Wq + (size_t)(h * DK_SZ) * K_SZ;
      v8f z = {};
      v8f acc = tile_gemm<64>(lane,
          [&](int m, int k) { return qin[(mb + m) * 65 + k]; },
          [&](int k, int n) { return WqH[(eb + n) * K_SZ + k]; }, z);
      tile_store(lane, acc, [&](int m, int n, float v) { qh[(mb + m) * 33 + eb + n] = v; });
    }
    // ---- key_h[d][n] = Wk_h (32x64) @ kf_h^T  (32x64), 8 tiles, 1/wave ----
    {
      const int db = (wv >> 2) << 4;
      const int nb = (wv & 3) << 4;
      const float* WkH = Wk + (size_t)h * DK_SZ * K_SZ;
      v8f z = {};
      v8f acc = tile_gemm<64>(lane,
          [&](int m, int k) { return WkH[(db + m) * K_SZ + k]; },
          [&](int k, int n) { return kf[(nb + n) * 65 + k]; }, z);
      tile_store(lane, acc, [&](int m, int n, float v) { keyb[(db + m) * 64 + nb + n] = v; });
    }
    __syncthreads();

    // ---- V = kf + sum_d relu(conv3(kf)) * Wv2  (in-place, sliding window) ----
    if (t < 64) {
      float w1[DV_SZ][KER_SZ], w2[DV_SZ];
#pragma unroll
      for (int d = 0; d < DV_SZ; ++d) {
#pragma unroll
        for (int kk = 0; kk < KER_SZ; ++kk) w1[d][kk] = Wv1[(h * DV_SZ + d) * KER_SZ + kk];
        w2[d] = Wv2[h * DV_SZ + d];
      }
      float* row = kf + t * 65;
      float a0 = row[0], a1 = row[1], a2;
      for (int p = 0; p < 64; ++p) {
        a2 = (p + 2 < 64) ? row[p + 2] : 0.0f;
        float s = 0.0f;
#pragma unroll
        for (int d = 0; d < DV_SZ; ++d) {
          float v1 = a0 * w1[d][0] + a1 * w1[d][1] + a2 * w1[d][2];
          s += fmaxf(v1, 0.0f) * w2[d];
        }
        row[p] = a0 + s;
        a0 = a1; a1 = a2;
      }
    }
    __syncthreads();

    // ---- scores = (q_h @ key) / sqrt(M)  (64x64, K=32), 16 tiles, 2/wave ----
#pragma unroll
    for (int j = 0; j < 2; ++j) {
      int ti = wv * 2 + j;
      int mb = (ti >> 2) << 4, nb = (ti & 3) << 4;
      v8f z = {};
      v8f acc = tile_gemm<32>(lane,
          [&](int m, int k) { return qh[(mb + m) * 33 + k]; },
          [&](int k, int n) { return keyb[k * 64 + nb + n]; }, z);
      tile_store(lane, acc,
          [&](int m, int n, float v) { sbuf[(mb + m) * 65 + nb + n] = v * 0.015625f; });
    }
    __syncthreads();

    // ---- row softmax ----
    if (t < 64) {
      float* row = sbuf + t * 65;
      float mx = row[0];
      for (int i = 1; i < 64; ++i) mx = fmaxf(mx, row[i]);
      float sum = 0.0f;
      for (int i = 0; i < 64; ++i) { float e = __expf(row[i] - mx); row[i] = e; sum += e; }
      float inv = 1.0f / sum;
      for (int i = 0; i < 64; ++i) row[i] *= inv;
    }
    __syncthreads();

    // ---- stream attn to global, compute O = attn @ V (K=64) ----
    float* attnOut = out + (size_t)B_SZ * M_SZ + (size_t)(b * H_SZ + h) * N_SZ * N_SZ;
#pragma unroll
    for (int i = 0; i < 16; ++i) {
      int j = t + i * 256;
      attnOut[j] = sbuf[(j >> 6) * 65 + (j & 63)];
    }
    v8f oacc[2];
#pragma unroll
    for (int j = 0; j < 2; ++j) {
      int ti = wv * 2 + j;
      int mb = (ti >> 2) << 4, nb = (ti & 3) << 4;
      v8f z = {};
      oacc[j] = tile_gemm<64>(lane,
          [&](int m, int k) { return sbuf[(mb + m) * 65 + k]; },
          [&](int k, int n) { return kf[k * 65 + nb + n]; }, z);
    }
    __syncthreads();  // all V reads complete before O overwrites kf
#pragma unroll
    for (int j = 0; j < 2; ++j) {
      int ti = wv * 2 + j;
      int mb = (ti >> 2) << 4, nb = (ti & 3) << 4;
      tile_store(lane, oacc[j], [&](int m, int n, float v) { kf[(mb + m) * 65 + nb + n] = v; });
    }
    __syncthreads();

    // ---- FC partial: out[n][o] += O_h[n][k] * Wfc[o][h*64+k]  (K=64) ----
    {
      int ti = wv * 2 + 0;
      int mb = (ti >> 2) << 4, ob = (ti & 3) << 4;
      facc0 = tile_gemm<64>(lane,
          [&](int m, int k) { return kf[(mb + m) * 65 + k]; },
          [&](int k, int n) { return Wfc[(ob + n) * (H_SZ * K_SZ) + h * 64 + k]; }, facc0);
    }
    {
      int ti = wv * 2 + 1;
      int mb = (ti >> 2) << 4, ob = (ti & 3) << 4;
      facc1 = tile_gemm<64>(lane,
          [&](int m, int k) { return kf[(mb + m) * 65 + k]; },
          [&](int k, int n) { return Wfc[(ob + n) * (H_SZ * K_SZ) + h * 64 + k]; }, facc1);
    }
  }

  // ---- store FC result: out[b][n*64 + o] ----
  {
    int ti = wv * 2 + 0;
    int mb = (ti >> 2) << 4, ob = (ti & 3) << 4;
    tile_store(lane, facc0, [&](int m, int n, float v) {
      out[(size_t)b * M_SZ + (mb + m) * 64 + ob + n] = v;
    });
  }
  {
    int ti = wv * 2 + 1;
    int mb = (ti >> 2) << 4, ob = (ti & 3) << 4;
    tile_store(lane, facc1, [&](int m, int n, float v) {
      out[(size_t)b * M_SZ + (mb + m) * 64 + ob + n] = v;
    });
  }
}

extern "C" void kernel_launch(void* const* d_in, const int* in_sizes, int n_in,
                              void* d_out, int out_size, void* d_ws, size_t ws_size,
                              hipStream_t stream) {
  (void)in_sizes; (void)n_in; (void)out_size; (void)d_ws; (void)ws_size;
  const float* features = (const float*)d_in[0];
  const int*   sidx     = (const int*)  d_in[1];
  const float* gamma    = (const float*)d_in[2];
  const float* beta     = (const float*)d_in[3];
  const float* Wq       = (const float*)d_in[4];
  const float* Wk       = (const float*)d_in[5];
  const float* Wv1      = (const float*)d_in[6];
  const float* Wv2      = (const float*)d_in[7];
  const float* Wfc      = (const float*)d_in[8];
  float* out = (float*)d_out;

  const size_t smem_bytes = (size_t)SMEM_FLOATS * sizeof(float);  // ~83 KB
  fused_attn_kernel<<<dim3(B_SZ), dim3(256), smem_bytes, stream>>>(
      features, sidx, gamma, beta, Wq, Wk, Wv1, Wv2, Wfc, out);
}